// MambaBlock_38826504356700
// MI455X (gfx1250) — compile-verified
//
#include <hip/hip_runtime.h>
#include <stdint.h>

// ---------------------------------------------------------------------------
// Mamba block for MI455X (gfx1250, wave32).
//  - GEMMs: V_WMMA_F32_16X16X4_F32, one wave = one 16x16 fp32 C tile.
//  - Selective scan: state in registers, B_t/C_t staged memory->LDS with
//    GLOBAL_LOAD_ASYNC_TO_LDS_B64 (ASYNCcnt), 3-buffer ring, depth-2 prefetch.
// ---------------------------------------------------------------------------

typedef float v2f __attribute__((ext_vector_type(2)));
typedef float v8f __attribute__((ext_vector_type(8)));

#define NUM_LAYERS 2
#define D_MODEL    64
#define D_INNER    128
#define D_STATE    256
#define D_CONV     4
#define DT_RANK    4
#define HIDDEN     128
#define BATCH      8
#define SEQ        512
#define ROWS       (BATCH * SEQ)   // 4096

__device__ __forceinline__ v8f wmma_f32_4(v2f a, v2f b, v8f c) {
  // D = A(16x4,f32) * B(4x16,f32) + C(16x16,f32)
  return __builtin_amdgcn_wmma_f32_16x16x4_f32(
      /*neg_a=*/false, a, /*neg_b=*/false, b,
      /*c_mod=*/(short)0, c, /*reuse_a=*/false, /*reuse_b=*/false);
}

// ---------------------------------------------------------------------------
// Generic fp32 WMMA GEMM:  C[M=4096][N] = A[M][K] * W[N][K]^T  (+ epilogue)
//   mode 0: plain   mode 1: +bias[n]   mode 2: softplus(acc + bias[n])
//
// f32 operand layout (CDNA5 ISA 7.12.2):
//   A 16x4 : lanes 0-15 VGPR{0,1}=K{0,1}; lanes 16-31 VGPR{0,1}=K{2,3}
//   B 4x16 : mirrored (lane = column, K pair by lane>>4)
//   C/D    : VGPR i -> row m0+i (lanes 0-15) / m0+8+i (lanes 16-31)
// ---------------------------------------------------------------------------
template <int K, int MODE>
__global__ void gemm_wmma_f32(const float* __restrict__ A,
                              const float* __restrict__ W,
                              float* __restrict__ C,
                              const float* __restrict__ bias,
                              int N, int ntiles_n) {
  const int wave = threadIdx.x >> 5;
  const int lane = threadIdx.x & 31;
  const int tile = blockIdx.x * (blockDim.x >> 5) + wave;
  const int tm = tile / ntiles_n;
  const int tn = tile - tm * ntiles_n;
  const int m0 = tm * 16;
  const int n0 = tn * 16;
  const int l16 = lane & 15;
  const int khalf = (lane >> 4) * 2;

  const float* __restrict__ Arow = A + (size_t)(m0 + l16) * K + khalf;
  const float* __restrict__ Wrow = W + (size_t)(n0 + l16) * K + khalf;

  v8f acc = {};
#pragma unroll
  for (int k = 0; k < K; k += 4) {
    v2f av = *(const v2f*)(Arow + k);
    v2f bv = *(const v2f*)(Wrow + k);
    acc = wmma_f32_4(av, bv, acc);
  }

  const int mbase = m0 + (lane >> 4) * 8;
  const int col = n0 + l16;
#pragma unroll
  for (int i = 0; i < 8; ++i) {
    float v = acc[i];
    if (MODE == 1) v += bias[col];
    if (MODE == 2) {
      v += bias[col];
      v = (v > 20.f) ? v : __logf(1.f + __expf(v));  // softplus
    }
    C[(size_t)(mbase + i) * N + col] = v;
  }
}

// ---------------------------------------------------------------------------
// Depthwise causal conv(4) + bias + SiLU.
// ---------------------------------------------------------------------------
__global__ void conv_silu_kernel(const float* __restrict__ xz,
                                 const float* __restrict__ w,
                                 const float* __restrict__ b,
                                 float* __restrict__ out) {
  const int idx = blockIdx.x * blockDim.x + threadIdx.x;  // ROWS*D_INNER
  const int d = idx & (D_INNER - 1);
  const int row = idx >> 7;
  const int lpos = row & (SEQ - 1);
  float acc = b[d];
#pragma unroll
  for (int k = 0; k < D_CONV; ++k) {
    const int ls = lpos - (D_CONV - 1) + k;
    const float v = (ls >= 0) ? xz[(size_t)(row - lpos + ls) * 256 + d] : 0.f;
    acc = fmaf(w[d * D_CONV + k], v, acc);
  }
  out[idx] = acc / (1.f + __expf(-acc));  // silu
}

// ---------------------------------------------------------------------------
// Fold dt_proj (128x4) with x_proj dt rows (4x128): W_dt[d][j] = sum_r dpw*xpw
// ---------------------------------------------------------------------------
__global__ void wdt_kernel(const float* __restrict__ dpw,
                           const float* __restrict__ xpw_dt,
                           float* __restrict__ wdt) {
  const int idx = blockIdx.x * blockDim.x + threadIdx.x;  // 128*128
  const int j = idx & (D_INNER - 1);
  const int d = idx >> 7;
  float s = 0.f;
#pragma unroll
  for (int r = 0; r < DT_RANK; ++r)
    s = fmaf(dpw[d * DT_RANK + r], xpw_dt[r * D_INNER + j], s);
  wdt[idx] = s;
}

// ---------------------------------------------------------------------------
// Selective scan. 128 blocks x 256 threads: block -> (batch, 8 channels),
// wave -> one d, lane -> 8 states s = lane + 32*j (conflict-free LDS banks).
//
// B_t/C_t (2KB/step) copied memory->LDS with GLOBAL_LOAD_ASYNC_TO_LDS_B64
// into a 3-slot ring; steps l+1 and l+2 stay in flight while step l computes.
// One s_wait_asynccnt + one __syncthreads per step (barrier also fences the
// ring-slot overwrite: slot (l+2)%3 was last read at step l-1, and every wave
// passed that read before the barrier at step l).
// Per-step scalars (dt,u,z) are software-prefetched one step ahead.
// ---------------------------------------------------------------------------
__global__ void scan_kernel(const float* __restrict__ u,
                            const float* __restrict__ dt,
                            const float* __restrict__ bc,
                            const float* __restrict__ xz,   // z = cols 128..255
                            const float* __restrict__ A_log,
                            const float* __restrict__ Dp,
                            float* __restrict__ out) {
  __shared__ float sBC[3][2 * D_STATE];  // 3 x 2KB ring

  const int tid = threadIdx.x;
  const int wave = tid >> 5;
  const int lane = tid & 31;
  const int bb = blockIdx.x >> 4;                 // batch 0..7
  const int d = ((blockIdx.x & 15) << 3) + wave;  // channel 0..127

  float Areg[8], h[8];
#pragma unroll
  for (int j = 0; j < 8; ++j) {
    Areg[j] = -__expf(A_log[(size_t)d * D_STATE + lane + 32 * j]);
    h[j] = 0.f;
  }
  const float dval = Dp[d];
  const size_t row0 = (size_t)bb * SEQ;

  // Each thread copies 8B per step; LDS dest offsets for the 3 ring slots.
  // (Generic LDS pointer truncated to 32 bits == LDS address, per flat rules.)
  uint32_t lds_slot[3];
#pragma unroll
  for (int s = 0; s < 3; ++s)
    lds_slot[s] = (uint32_t)(uintptr_t)&sBC[s][2 * tid];

  const float* gsrc0 = bc + row0 * (2 * D_STATE) + 2 * tid;

#define ISSUE_ASYNC(step, slot)                                              \
  do {                                                                       \
    const float* _g = gsrc0 + (size_t)(step) * (2 * D_STATE);                \
    asm volatile("global_load_async_to_lds_b64 %0, %1, off"                  \
                 :: "v"(lds_slot[slot]), "v"(_g) : "memory");                \
  } while (0)

  // Prime the pipeline: steps 0 and 1 in flight.
  ISSUE_ASYNC(0, 0);
  ISSUE_ASYNC(1, 1);

  // Prefetched scalars for step 0.
  float dtv = dt[row0 * D_INNER + d];
  float uv  = u[row0 * D_INNER + d];
  float zv  = xz[row0 * 256 + D_INNER + d];

  int cur = 0;  // ring slot holding step l
  for (int l = 0; l < SEQ; ++l) {
    // Own copy of step l finished (async ops complete in order).
    if (l + 1 < SEQ) asm volatile("s_wait_asynccnt 0x1" ::: "memory");
    else             asm volatile("s_wait_asynccnt 0x0" ::: "memory");
    __syncthreads();  // all waves' step-l data visible; slot reuse fenced

    // Keep the pipeline 2 deep.
    const int tgt = (cur + 2 >= 3) ? cur - 1 : cur + 2;
    if (l + 2 < SEQ) ISSUE_ASYNC(l + 2, tgt);

    // Prefetch next step's scalars (latency overlaps compute below).
    float dtn = 0.f, un = 0.f, zn = 0.f;
    if (l + 1 < SEQ) {
      const size_t rown = row0 + l + 1;
      dtn = dt[rown * D_INNER + d];
      un  = u[rown * D_INNER + d];
      zn  = xz[rown * 256 + D_INNER + d];
    }

    const float* __restrict__ bufp = sBC[cur];
    float acc = 0.f;
#pragma unroll
    for (int j = 0; j < 8; ++j) {
      const int s = lane + 32 * j;
      const float dA = __expf(dtv * Areg[j]);
      h[j] = fmaf(h[j], dA, dtv * bufp[s] * uv);
      acc = fmaf(h[j], bufp[D_STATE + s], acc);
    }
    // wave32 all-reduce
#pragma unroll
    for (int off = 16; off >= 1; off >>= 1)
      acc += __shfl_xor(acc, off, 32);

    if (lane == 0) {
      const float yv = fmaf(uv, dval, acc);
      out[(row0 + l) * D_INNER + d] = yv * (zv / (1.f + __expf(-zv)));
    }

    dtv = dtn; uv = un; zv = zn;
    cur = (cur + 1 == 3) ? 0 : cur + 1;
  }
#undef ISSUE_ASYNC
}

// ---------------------------------------------------------------------------
// Launch
// ---------------------------------------------------------------------------
extern "C" void kernel_launch(void* const* d_in, const int* in_sizes, int n_in,
                              void* d_out, int out_size, void* d_ws, size_t ws_size,
                              hipStream_t stream) {
  (void)in_sizes; (void)n_in; (void)out_size; (void)ws_size;
  const float* x     = (const float*)d_in[0];
  const float* ipw   = (const float*)d_in[1];
  const float* cw    = (const float*)d_in[2];
  const float* cb    = (const float*)d_in[3];
  const float* xpw   = (const float*)d_in[4];
  const float* dpw   = (const float*)d_in[5];
  const float* dpb   = (const float*)d_in[6];
  const float* A_log = (const float*)d_in[7];
  const float* Dp    = (const float*)d_in[8];
  const float* opw   = (const float*)d_in[9];
  const float* fcw   = (const float*)d_in[10];
  const float* fcb   = (const float*)d_in[11];

  float* ws   = (float*)d_ws;
  float* xz   = ws;                          // ROWS*256
  float* ucv  = xz  + (size_t)ROWS * 256;    // ROWS*128
  float* bc   = ucv + (size_t)ROWS * 128;    // ROWS*512
  float* dt   = bc  + (size_t)ROWS * 512;    // ROWS*128
  float* wdt  = dt  + (size_t)ROWS * 128;    // 128*128
  float* yz   = wdt + 128 * 128;             // ROWS*128
  float* hbuf = yz  + (size_t)ROWS * 128;    // ROWS*64

  const int MT = ROWS / 16;  // 256 M-tiles
  const float* layer_in = x;

  for (int l = 0; l < NUM_LAYERS; ++l) {
    const float* ipw_l = ipw + (size_t)l * (2 * D_INNER) * D_MODEL;
    const float* cw_l  = cw  + (size_t)l * D_INNER * D_CONV;
    const float* cb_l  = cb  + (size_t)l * D_INNER;
    const float* xpw_l = xpw + (size_t)l * (DT_RANK + 2 * D_STATE) * D_INNER;
    const float* dpw_l = dpw + (size_t)l * D_INNER * DT_RANK;
    const float* dpb_l = dpb + (size_t)l * D_INNER;
    const float* Al_l  = A_log + (size_t)l * D_INNER * D_STATE;
    const float* Dp_l  = Dp  + (size_t)l * D_INNER;
    const float* opw_l = opw + (size_t)l * D_MODEL * D_INNER;

    // 1) xz = layer_in @ ipw^T : (ROWS,256), K=64
    gemm_wmma_f32<64, 0><<<MT * 16 / 8, 256, 0, stream>>>(
        layer_in, ipw_l, xz, nullptr, 256, 16);

    // 2) depthwise causal conv + SiLU -> u
    conv_silu_kernel<<<ROWS * D_INNER / 256, 256, 0, stream>>>(xz, cw_l, cb_l, ucv);

    // 3) W_dt = dpw @ xpw[0:4]   (128x128)
    wdt_kernel<<<D_INNER * D_INNER / 256, 256, 0, stream>>>(dpw_l, xpw_l, wdt);

    // 4) BC = u @ xpw[4:516]^T : (ROWS,512), K=128
    gemm_wmma_f32<128, 0><<<MT * 32 / 8, 256, 0, stream>>>(
        ucv, xpw_l + DT_RANK * D_INNER, bc, nullptr, 512, 32);

    // 5) dt = softplus(u @ W_dt^T + dpb) : (ROWS,128), K=128
    gemm_wmma_f32<128, 2><<<MT * 8 / 8, 256, 0, stream>>>(
        ucv, wdt, dt, dpb_l, 128, 8);

    // 6) selective scan (+ *silu(z) + u*Dp epilogue)
    scan_kernel<<<BATCH * 16, 256, 0, stream>>>(ucv, dt, bc, xz, Al_l, Dp_l, yz);

    // 7) out = yz @ opw^T : (ROWS,64), K=128
    gemm_wmma_f32<128, 0><<<MT * 4 / 8, 256, 0, stream>>>(
        yz, opw_l, hbuf, nullptr, 64, 4);

    layer_in = hbuf;
  }

  // final fc: out = h @ fcw^T + fcb : (ROWS,128), K=64
  gemm_wmma_f32<64, 1><<<MT * 8 / 8, 256, 0, stream>>>(
      hbuf, fcw, (float*)d_out, fcb, 128, 8);
}